// block_DCT_24352464569478
// MI455X (gfx1250) — compile-verified
//
#include <hip/hip_runtime.h>

// fp32 WMMA fragment types for V_WMMA_F32_16X16X4_F32 (wave32):
//  A: 16x4 f32  -> 2 VGPRs/lane,  C/D: 16x16 f32 -> 8 VGPRs/lane
typedef __attribute__((ext_vector_type(2))) float v2f;
typedef __attribute__((ext_vector_type(8))) float v8f;

#define IMG    224
#define WS     8
#define NPATCH 109          // (224-8)/2 + 1
#define NQUAD  55           // ceil(109/2)
#define BC     48           // 16 batches * 3 channels

// D = A x B + C, one K=4 slice; chain 4 of these for a 16x16x16 product.
#define WMMA4(A, B, C) \
    __builtin_amdgcn_wmma_f32_16x16x4_f32(false, (A), false, (B), (short)0, (C), false, false)

__launch_bounds__(256)
__global__ void dct_band_filter_kernel(const float* __restrict__ x,
                                       const float* __restrict__ Dm,
                                       const float* __restrict__ filt,
                                       float* __restrict__ out)
{
    // Per-block constants + per-wave private slices (8 waves / block).
    __shared__ float sD[64];
    __shared__ float sF[64];
    __shared__ float sIn [8][10 * 12];   // 10x10 input footprint (padded stride 12)
    __shared__ float sAcc[8][10 * 12];   // 10x10 overlap-add accumulator
    __shared__ float sTile[8][16 * 17];  // 16x16 fragment round-trip (pad 17 vs banks)

    const int tid  = threadIdx.x;
    const int lane = tid & 31;
    const int wave = tid >> 5;

    if (tid < 64)        sD[tid]      = Dm[tid];
    else if (tid < 128)  sF[tid - 64] = filt[tid - 64];

    // One wave == one 2x2 quad of stride-2 patches.
    const int quadId = blockIdx.x * 8 + wave;
    const int qn = quadId % NQUAD;
    const int qm = (quadId / NQUAD) % NQUAD;
    const int bc = quadId / (NQUAD * NQUAD);

    const int pm0 = qm * 2, pn0 = qn * 2;      // patch indices of quad corner
    const int h0  = pm0 * 2, w0 = pn0 * 2;     // pixel origin of 10x10 footprint
    const float* __restrict__ xin = x   + (size_t)bc * (IMG * IMG);
    float*       __restrict__ po  = out + (size_t)bc * (IMG * IMG);

    // Phase 1: stage the 10x10 footprint in LDS, zero the accumulator.
    #pragma unroll
    for (int t = 0; t < 4; ++t) {
        int idx = lane + 32 * t;
        if (idx < 100) {
            int r = idx / 10, c = idx % 10;
            int h = h0 + r,  w = w0 + c;
            sIn [wave][r * 12 + c] = (h < IMG && w < IMG) ? xin[h * IMG + w] : 0.0f;
            sAcc[wave][r * 12 + c] = 0.0f;
        }
    }
    __syncthreads();

    const int half = lane >> 4;   // K-half for A/B frags, row-half for C/D frags
    const int nc   = lane & 15;   // A-frag row / B-frag column / C-frag column

    // Constant A fragments (ISA 16x4 A layout: VGPR v holds K = 4t + 2*half + v, M = nc).
    //  aF = blockdiag(D, D)      (forward DCT stages)
    //  aI = blockdiag(D^T, D^T)  (inverse DCT stages)
    v2f aF[4], aI[4];
    #pragma unroll
    for (int t = 0; t < 4; ++t) {
        #pragma unroll
        for (int v = 0; v < 2; ++v) {
            int k = 4 * t + 2 * half + v;
            float f = 0.0f, g = 0.0f;
            if ((nc >> 3) == (k >> 3)) {
                f = sD[(nc & 7) * 8 + (k & 7)];
                g = sD[(k & 7) * 8 + (nc & 7)];
            }
            aF[t][v] = f;
            aI[t][v] = g;
        }
    }
    // Mask registers: in the C/D layout row&7 == VGPR index, col == nc (filt symmetric).
    float ff[8];
    #pragma unroll
    for (int v = 0; v < 8; ++v) ff[v] = sF[v * 8 + (nc & 7)];

    // Phase 2: pack the 2x2 patch quad into B fragments (quadrant (qr,qc) = patch).
    // Assumed B layout mirrors A: VGPR v holds K = 4t + 2*half + v, N = nc.
    const int colIn = 2 * (nc >> 3) + (nc & 7);  // footprint column for this lane
    v2f bP[4];
    #pragma unroll
    for (int t = 0; t < 4; ++t) {
        #pragma unroll
        for (int v = 0; v < 2; ++v) {
            int K = 4 * t + 2 * half + v;
            int rowIn = 2 * (K >> 3) + (K & 7);
            bP[t][v] = sIn[wave][rowIn * 12 + colIn];
        }
    }

    // Stage 1: C1 = blockdiag(D) * P        (quadrants: D*P_ab)
    v8f c1 = {};
    #pragma unroll
    for (int t = 0; t < 4; ++t) c1 = WMMA4(aF[t], bP[t], c1);

    // Transpose via LDS, Stage 2: C2 = blockdiag(D) * C1^T  -> quadrant (a,b) = Xd_ba^T
    #pragma unroll
    for (int v = 0; v < 8; ++v) sTile[wave][(v + 8 * half) * 17 + nc] = c1[v];
    __syncthreads();
    v2f b2[4];
    #pragma unroll
    for (int t = 0; t < 4; ++t) {
        #pragma unroll
        for (int v = 0; v < 2; ++v) {
            int K = 4 * t + 2 * half + v;
            b2[t][v] = sTile[wave][nc * 17 + K];        // transposed read
        }
    }
    __syncthreads();
    v8f c2 = {};
    #pragma unroll
    for (int t = 0; t < 4; ++t) c2 = WMMA4(aF[t], b2[t], c2);

    // Frequency-band mask (filt symmetric -> commutes with the transposes).
    #pragma unroll
    for (int v = 0; v < 8; ++v) c2[v] *= ff[v];

    // Stage 3: C3 = blockdiag(D^T) * C2 (straight relayout C/D -> B through LDS)
    #pragma unroll
    for (int v = 0; v < 8; ++v) sTile[wave][(v + 8 * half) * 17 + nc] = c2[v];
    __syncthreads();
    v2f b3[4];
    #pragma unroll
    for (int t = 0; t < 4; ++t) {
        #pragma unroll
        for (int v = 0; v < 2; ++v) {
            int K = 4 * t + 2 * half + v;
            b3[t][v] = sTile[wave][K * 17 + nc];        // non-transposed read
        }
    }
    __syncthreads();
    v8f c3 = {};
    #pragma unroll
    for (int t = 0; t < 4; ++t) c3 = WMMA4(aI[t], b3[t], c3);

    // Stage 4: C4 = blockdiag(D^T) * C3^T -> quadrant (a,b) = Y_ab (patch result)
    #pragma unroll
    for (int v = 0; v < 8; ++v) sTile[wave][(v + 8 * half) * 17 + nc] = c3[v];
    __syncthreads();
    v2f b4[4];
    #pragma unroll
    for (int t = 0; t < 4; ++t) {
        #pragma unroll
        for (int v = 0; v < 2; ++v) {
            int K = 4 * t + 2 * half + v;
            b4[t][v] = sTile[wave][nc * 17 + K];        // transposed read
        }
    }
    __syncthreads();
    v8f c4 = {};
    #pragma unroll
    for (int t = 0; t < 4; ++t) c4 = WMMA4(aI[t], b4[t], c4);

    // Phase 3: combine the 4 overlapping patch outputs into the 10x10 accumulator.
    const int qc = nc >> 3;                 // quadrant column == patch dn
    const int accCol = colIn;               // same footprint column as the input
    const bool colValid = (pn0 + qc) < NPATCH;
    #pragma unroll
    for (int v = 0; v < 8; ++v) {
        // C/D layout: row = v + 8*half -> quadrant row = half, in-patch row i = v
        int accRow = 2 * half + v;
        if (colValid && (pm0 + half) < NPATCH)
            atomicAdd(&sAcc[wave][accRow * 12 + accCol], c4[v]);
    }
    __syncthreads();

    // Phase 4: overlap-add into the image (quads overlap each other -> atomics).
    #pragma unroll
    for (int t = 0; t < 4; ++t) {
        int idx = lane + 32 * t;
        if (idx < 100) {
            int r = idx / 10, c = idx % 10;
            int h = h0 + r,  w = w0 + c;
            if (h < IMG && w < IMG)
                atomicAdd(&po[h * IMG + w], sAcc[wave][r * 12 + c]);
        }
    }
}

extern "C" void kernel_launch(void* const* d_in, const int* in_sizes, int n_in,
                              void* d_out, int out_size, void* d_ws, size_t ws_size,
                              hipStream_t stream) {
    (void)in_sizes; (void)n_in; (void)d_ws; (void)ws_size;
    const float* x    = (const float*)d_in[0];
    const float* D    = (const float*)d_in[1];
    const float* filt = (const float*)d_in[2];
    float* out = (float*)d_out;

    // Overlap-add target must start at zero every call (buffer is poisoned).
    hipMemsetAsync(out, 0, (size_t)out_size * sizeof(float), stream);

    const int nQuads  = BC * NQUAD * NQUAD;   // 145200, divisible by 8 waves/block
    const int nBlocks = nQuads / 8;           // 18150
    dct_band_filter_kernel<<<nBlocks, 256, 0, stream>>>(x, D, filt, out);
}